// StochasticPool2d_84602265797119
// MI455X (gfx1250) — compile-verified
//
#include <hip/hip_runtime.h>
#include <stdint.h>

// Problem constants (from reference): x (16,96,224,224) f32 -> out (16,96,112,112) f32
#define B_  16
#define C_  96
#define H_  224
#define W_  224
#define OH_ 112
#define OW_ 112

#define TILE_OROWS  4                     // output rows per TDM tile
#define TILE_IROWS  (TILE_OROWS * 2)      // 8 input rows
#define TILE_FLOATS (TILE_IROWS * W_)     // 1792 floats, fully contiguous (whole rows)
#define TILE_BYTES  (TILE_FLOATS * 4)     // 7168 B
#define NTILES      (OH_ / TILE_OROWS)    // 28 tiles per (b,c) plane
#define NTHREADS    224                   // 7 wave32's; 448 outputs/tile = 2 per thread

typedef __attribute__((ext_vector_type(4))) unsigned int u32x4;
typedef __attribute__((ext_vector_type(8))) unsigned int u32x8;

// Deterministic counter-based RNG (PCG-style hash of the output linear index).
__device__ __forceinline__ float rnd01(uint32_t idx) {
  uint32_t h = idx * 747796405u + 2891336453u;
  h = ((h >> ((h >> 28) + 4u)) ^ h) * 277803737u;
  h = (h >> 22) ^ h;
  return (float)(h >> 8) * (1.0f / 16777216.0f);   // [0,1)
}

// Softmax over the 4 window values + inverse-CDF categorical sample.
__device__ __forceinline__ float stochastic_pick(float v0, float v1, float v2,
                                                 float v3, uint32_t idx) {
  float m  = fmaxf(fmaxf(v0, v1), fmaxf(v2, v3));
  float e0 = __expf(v0 - m);
  float e1 = __expf(v1 - m);
  float e2 = __expf(v2 - m);
  float e3 = __expf(v3 - m);
  float c0 = e0, c1 = c0 + e1, c2 = c1 + e2, s = c2 + e3;
  float u  = rnd01(idx) * s;
  return (u < c0) ? v0 : (u < c1) ? v1 : (u < c2) ? v2 : v3;
}

// Issue a TDM 1-D tile DMA: nfloats contiguous f32 from global `gsrc`
// into LDS at `lds_dst`.  D# layout per CDNA5 ISA 8.3/8.4.
// Groups 2/3 omitted -> tensor treated as <=2D.  Tracked by TENSORcnt.
//
// IMPORTANT: lds_dst is passed as an asm input so the LDS buffer *escapes*;
// combined with the "memory" clobber this tells LLVM the asm may write the
// shared array (otherwise the subsequent ds_load's fold to undef and the
// whole compute chain is dead-code eliminated, as seen in round 1).
__device__ __forceinline__ void tdm_load_1d(const float* gsrc, float4* lds_dst,
                                            uint32_t nfloats) {
  const uint64_t gaddr = (uint64_t)(uintptr_t)gsrc;
  // Low 32 bits of the flat address of an LDS object == LDS byte offset
  // (aperture occupies addr[63:32]).
  const uint32_t lds_off = (uint32_t)(uintptr_t)lds_dst;
  u32x4 g0;
  g0.x = 1u;                                      // [1:0] count=1 (valid descriptor)
  g0.y = lds_off;                                 // [63:32] lds_addr (bytes)
  g0.z = (uint32_t)gaddr;                         // [95:64] global_addr lo
  g0.w = ((uint32_t)(gaddr >> 32) & 0x01FFFFFFu)  // [120:96] global_addr hi
         | (2u << 30);                            // [127:126] type=2 ("image")
  u32x8 g1;
  g1.s0 = (2u << 16);                             // data_size=2 -> 4 bytes; wg_mask=0
  g1.s1 = (nfloats & 0xFFFFu) << 16;              // tensor_dim0[15:0] @ bits 63:48
  g1.s2 = (nfloats >> 16) | (1u << 16);           // tensor_dim0[31:16] | tensor_dim1=1
  g1.s3 = (nfloats & 0xFFFFu) << 16;              // tile_dim0 @ bits 127:112 (<=65535)
  g1.s4 = 1u;                                     // tile_dim1=1, tile_dim2=0
  g1.s5 = nfloats;                                // tensor_dim0_stride lo32
  g1.s6 = 0u;                                     // stride0 hi | stride1 lo
  g1.s7 = 0u;                                     // stride1 hi
  asm volatile("tensor_load_to_lds %0, %1"
               :
               : "s"(g0), "s"(g1), "r"(lds_dst)   // lds_dst: escape only
               : "memory");
}

__global__ __launch_bounds__(NTHREADS)
void StochasticPool2d_kernel(const float* __restrict__ x, float* __restrict__ out) {
  // Double-buffered tile store.
  __shared__ float4 lds[2][TILE_FLOATS / 4];   // 2 x 7168 B

  const int      bc        = blockIdx.x;                    // (b*C + c)
  const size_t   plane_in  = (size_t)bc * (H_ * W_);
  const uint32_t plane_out = (uint32_t)bc * (OH_ * OW_);
  const float*   xp        = x + plane_in;
  float*         op        = out + plane_out;

  const int  tid    = threadIdx.x;
  const bool issuer = (tid < 32);              // wave 0 drives the TDM pipeline

  // Prologue: kick off tile 0 DMA.
  if (issuer) {
    tdm_load_1d(xp, &lds[0][0], TILE_FLOATS);
  }

  const int orow = tid / 56;                   // output row within tile: 0..3
  const int g    = tid % 56;                   // float4 group -> output cols 2g,2g+1

  for (int t = 0; t < NTILES; ++t) {
    const int buf = t & 1;

    if (issuer) {
      if (t + 1 < NTILES) {
        // Software pipeline: issue tile t+1 before waiting for tile t.
        tdm_load_1d(xp + (size_t)(t + 1) * TILE_FLOATS, &lds[(t + 1) & 1][0],
                    TILE_FLOATS);
        __builtin_amdgcn_s_wait_tensorcnt(1);  // in-order: tile t complete
      } else {
        __builtin_amdgcn_s_wait_tensorcnt(0);
      }
    }
    __syncthreads();                           // lds[buf] holds tile t for all waves

    // Two vertically adjacent input rows of this thread's two 2x2 windows.
    const float4 a = lds[buf][(2 * orow)     * (W_ / 4) + g];  // ds_load_b128
    const float4 b = lds[buf][(2 * orow + 1) * (W_ / 4) + g];  // ds_load_b128

    const int      oy  = t * TILE_OROWS + orow;
    const uint32_t lin = plane_out + (uint32_t)oy * OW_ + (uint32_t)(2 * g);

    // Window order matches reference: [top-left, top-right, bot-left, bot-right].
    float2 r;
    r.x = stochastic_pick(a.x, a.y, b.x, b.y, lin);
    r.y = stochastic_pick(a.z, a.w, b.z, b.w, lin + 1u);
    *(float2*)(op + (size_t)oy * OW_ + 2 * g) = r;             // global_store_b64

    __syncthreads();                           // protect lds[buf] before tile t+2 DMA
  }
}

extern "C" void kernel_launch(void* const* d_in, const int* in_sizes, int n_in,
                              void* d_out, int out_size, void* d_ws, size_t ws_size,
                              hipStream_t stream) {
  const float* x   = (const float*)d_in[0];
  float*       out = (float*)d_out;
  dim3 grid(B_ * C_);        // 1536 workgroups, one per (b,c) plane
  dim3 block(NTHREADS);      // 224 threads = 7 wave32's
  hipLaunchKernelGGL(StochasticPool2d_kernel, grid, block, 0, stream, x, out);
}